// CCNE_1967095021604
// MI455X (gfx1250) — compile-verified
//
#include <hip/hip_runtime.h>
#include <math.h>

typedef __attribute__((ext_vector_type(2))) float v2f;
typedef __attribute__((ext_vector_type(8))) float v8f;

#define D_IN 128   // input feature dim for both GEMMs (D_S = D_T = HID = 128)

// ---------------- elementwise helpers ----------------

__global__ void k_fill(float* __restrict__ p, float v, int n) {
  int i = blockIdx.x * blockDim.x + threadIdx.x;
  if (i < n) p[i] = v;
}

__global__ void k_count_deg(const int* __restrict__ dst, float* __restrict__ deg, int E) {
  int e = blockIdx.x * blockDim.x + threadIdx.x;
  if (e < E) atomicAdd(&deg[dst[e]], 1.0f);
}

// deg was accumulated into `invdeg`; produce dis = deg^-1/2 and invdeg = 1/deg in place
__global__ void k_deg_finish(float* __restrict__ invdeg, float* __restrict__ dis, int n) {
  int i = blockIdx.x * blockDim.x + threadIdx.x;
  if (i < n) {
    float d = invdeg[i];
    dis[i] = rsqrtf(d);
    invdeg[i] = 1.0f / d;
  }
}

// ---------------- fp32 WMMA GEMM: H[N,DOUT] = X[N,128] @ W[128,DOUT] ----------------
// One wave per 16x16 output tile; DOUT/16 waves per block cover a 16-row strip.
// A layout (16x4 f32): lanes 0-15 -> M=lane, {K+0,K+1}; lanes 16-31 -> M=lane-16, {K+2,K+3}
// B layout (4x16 f32): mirrored, N striped across lanes
// C/D layout (16x16 f32): VGPR r -> row r (lanes 0-15) / row r+8 (lanes 16-31), col = lane%16
template <int DOUT>
__global__ __launch_bounds__(32 * (DOUT / 16)) void k_gemm_f32_wmma(
    const float* __restrict__ X, const float* __restrict__ W, float* __restrict__ H) {
  const int lane = threadIdx.x;        // 0..31
  const int wy   = threadIdx.y;        // which 16-col tile
  const int row0 = blockIdx.x * 16;
  const int col0 = wy * 16;
  const int half = lane >> 4;          // 0: lanes 0-15, 1: lanes 16-31
  const int lm   = lane & 15;

  const float* xrow = X + (size_t)(row0 + lm) * D_IN;
  v8f c = {};

#pragma unroll
  for (int k = 0; k < D_IN; k += 4) {
    const int ka = k + 2 * half;
    v2f a = *(const v2f*)(xrow + ka);                 // contiguous 8B load
    v2f b;
    b.x = W[(size_t)ka * DOUT + col0 + lm];
    b.y = W[(size_t)(ka + 1) * DOUT + col0 + lm];
    c = __builtin_amdgcn_wmma_f32_16x16x4_f32(
        /*neg_a=*/false, a, /*neg_b=*/false, b,
        /*c_mod=*/(short)0, c, /*reuse_a=*/false, /*reuse_b=*/false);
  }

  const int col = col0 + lm;
#pragma unroll
  for (int r = 0; r < 8; ++r) {
    const int row = row0 + (half ? (r + 8) : r);
    H[(size_t)row * DOUT + col] = c[r];
  }
}

// ---------------- edge scatter: acc[dst] += dis[src] * h[src] ----------------
// One wave (32 lanes) per edge; D/32 floats per lane, float4/float2 gathers,
// scalar global_atomic_add_f32 scatters (L2-resident, 23.3 TB/s-class fabric).
template <int D>
__global__ void k_scatter(const float* __restrict__ h, const int* __restrict__ src,
                          const int* __restrict__ dst, const float* __restrict__ dis,
                          float* __restrict__ acc, int E) {
  int gid  = blockIdx.x * blockDim.x + threadIdx.x;
  int e    = gid >> 5;
  int lane = gid & 31;
  if (e >= E) return;
  int s = src[e], d = dst[e];
  float m = dis[s];
  if (D == 128) {
    float4 v = ((const float4*)(h + (size_t)s * D))[lane];
    float* a = acc + (size_t)d * D + lane * 4;
    atomicAdd(a + 0, v.x * m);
    atomicAdd(a + 1, v.y * m);
    atomicAdd(a + 2, v.z * m);
    atomicAdd(a + 3, v.w * m);
  } else {  // D == 64
    float2 v = ((const float2*)(h + (size_t)s * D))[lane];
    float* a = acc + (size_t)d * D + lane * 2;
    atomicAdd(a + 0, v.x * m);
    atomicAdd(a + 1, v.y * m);
  }
}

// ---------------- out = act(dis[n]*acc + h*invdeg[n] + b) ----------------
template <int D, bool RELU>
__global__ void k_finalize(const float* __restrict__ acc, const float* __restrict__ h,
                           const float* __restrict__ dis, const float* __restrict__ invdeg,
                           const float* __restrict__ bias, float* __restrict__ out, int n) {
  int gid = blockIdx.x * blockDim.x + threadIdx.x;
  if (gid >= n * D) return;
  int node = gid / D;
  int dcol = gid & (D - 1);
  float v = dis[node] * acc[gid] + h[gid] * invdeg[node] + bias[dcol];
  if (RELU) v = fmaxf(v, 0.0f);
  out[gid] = v;
}

// ---------------- decoder: p[e] = sigmoid(dot(z[src], z[dst])) over 64 dims ----------------
// 16 lanes per edge (2 edges per wave32), float4 loads, xor-shuffle reduction.
__global__ void k_decoder(const float* __restrict__ z, const int* __restrict__ src,
                          const int* __restrict__ dst, float* __restrict__ p, int E) {
  int gid  = blockIdx.x * blockDim.x + threadIdx.x;
  int e    = gid >> 4;
  int lane = gid & 15;
  if (e >= E) return;
  int s = src[e], d = dst[e];
  float4 a = ((const float4*)(z + (size_t)s * 64))[lane];
  float4 b = ((const float4*)(z + (size_t)d * 64))[lane];
  float part = a.x * b.x + a.y * b.y + a.z * b.z + a.w * b.w;
  part += __shfl_xor(part, 8, 32);
  part += __shfl_xor(part, 4, 32);
  part += __shfl_xor(part, 2, 32);
  part += __shfl_xor(part, 1, 32);
  if (lane == 0) p[e] = 1.0f / (1.0f + __expf(-part));
}

// ---------------- driver ----------------

extern "C" void kernel_launch(void* const* d_in, const int* in_sizes, int n_in,
                              void* d_out, int out_size, void* d_ws, size_t ws_size,
                              hipStream_t stream) {
  const float* xs = (const float*)d_in[0];
  const float* xt = (const float*)d_in[1];
  const int*  sEI = (const int*)d_in[2];
  const int*  tEI = (const int*)d_in[3];
  const float* W1 = (const float*)d_in[4];
  const float* b1 = (const float*)d_in[5];
  const float* W2 = (const float*)d_in[6];
  const float* b2 = (const float*)d_in[7];
  const float* W3 = (const float*)d_in[8];
  const float* b3 = (const float*)d_in[9];

  const int N = in_sizes[0] / 128;   // 100000 (multiple of 16)
  const int E = in_sizes[2] / 2;     // 500000
  const int D = 128, O = 64;

  // workspace layout (floats): h[N*128] | acc[N*128] (reused as h2[N*64]+acc2[N*64]) | dis[N] | inv[N]
  float* ws   = (float*)d_ws;
  float* h    = ws;
  float* acc  = ws + (size_t)N * D;
  float* h2   = acc;
  float* acc2 = acc + (size_t)N * O;
  float* dis  = ws + 2 * (size_t)N * D;
  float* inv  = dis + N;

  // output layout: zs[N*64] | zt[N*64] | ps[E] | pt[E]
  float* zs = (float*)d_out;
  float* zt = zs + (size_t)N * O;
  float* ps = zt + (size_t)N * O;
  float* pt = ps + E;

  auto run = [&](const float* x, const float* Wa, const float* ba,
                 const int* ei, float* z, float* p) {
    const int* srcp = ei;
    const int* dstp = ei + E;
    int nb;
    // degrees (with self-loop +1), dis = deg^-1/2, inv = 1/deg
    nb = (N + 255) / 256;      k_fill<<<nb, 256, 0, stream>>>(inv, 1.0f, N);
    nb = (E + 255) / 256;      k_count_deg<<<nb, 256, 0, stream>>>(dstp, inv, E);
    nb = (N + 255) / 256;      k_deg_finish<<<nb, 256, 0, stream>>>(inv, dis, N);
    // conv layer 1: h = x@Wa ; aggregate ; relu(+bias) in place over h
    k_gemm_f32_wmma<128><<<N / 16, dim3(32, 8), 0, stream>>>(x, Wa, h);
    nb = (N * D + 255) / 256;  k_fill<<<nb, 256, 0, stream>>>(acc, 0.0f, N * D);
    nb = (E * 32 + 255) / 256; k_scatter<128><<<nb, 256, 0, stream>>>(h, srcp, dstp, dis, acc, E);
    nb = (N * D + 255) / 256;  k_finalize<128, true><<<nb, 256, 0, stream>>>(acc, h, dis, inv, ba, h, N);
    // conv layer 2 (shared W3): h2 = h@W3 ; aggregate ; z = result + b3
    k_gemm_f32_wmma<64><<<N / 16, dim3(32, 4), 0, stream>>>(h, W3, h2);
    nb = (N * O + 255) / 256;  k_fill<<<nb, 256, 0, stream>>>(acc2, 0.0f, N * O);
    nb = (E * 32 + 255) / 256; k_scatter<64><<<nb, 256, 0, stream>>>(h2, srcp, dstp, dis, acc2, E);
    nb = (N * O + 255) / 256;  k_finalize<64, false><<<nb, 256, 0, stream>>>(acc2, h2, dis, inv, b3, z, N);
    // decoder
    nb = (E * 16 + 255) / 256; k_decoder<<<nb, 256, 0, stream>>>(z, srcp, dstp, p, E);
  };

  run(xs, W1, b1, sEI, zs, ps);
  run(xt, W2, b2, tEI, zt, pt);
}